// ChannelLatentMixer_48661979464238
// MI455X (gfx1250) — compile-verified
//
#include <hip/hip_runtime.h>

// ---------------------------------------------------------------------------
// ChannelLatentMixer for MI455X (gfx1250, wave32)
//
// out[b] = concat(z[b] (64x128), mean_{b': ch[b']==ch[b]} z[b'] )  -> (B,128,128)
//
// Bandwidth-bound op (~0.25 flop/byte): min traffic = read z once (134 MB) +
// write out (268 MB) ~= 402 MB ~= 17 us at 23.3 TB/s. Design:
//   - segment-sum expressed as one-hot GEMM on V_WMMA_F32_16X16X4_F32
//   - z read exactly ONCE (WMMA pass also writes the copy-half of out)
//   - 1 MB sums table stays resident in the 192 MB L2 for the broadcast pass
//   - channel-id staging into LDS via the Tensor Data Mover (TENSOR_LOAD_TO_LDS
//     + s_wait_tensorcnt) where the toolchain exposes it
// ---------------------------------------------------------------------------

typedef __attribute__((ext_vector_type(2))) float v2f;
typedef __attribute__((ext_vector_type(8))) float v8f;
typedef __attribute__((ext_vector_type(4))) unsigned int u32x4;
typedef __attribute__((ext_vector_type(8))) int i32x8;
typedef __attribute__((ext_vector_type(4))) int i32x4;

#define B_ROWS   4096
#define C_CH     32
#define ND       8192            // 64 * 128 elements per z row
#define OUT_ROW  16384           // 128 * 128 elements per out row
#define KSPLIT   8
#define BCHUNK   (B_ROWS / KSPLIT)   // 512 rows per K-split

#if defined(__gfx1250__) && __has_builtin(__builtin_amdgcn_tensor_load_to_lds)
#define HAVE_TDM 1
#else
#define HAVE_TDM 0
#endif

// ---- zero scratch (inv_count[32] + sums[32][8192]) ------------------------
__global__ void k_zero(float* ws, int n) {
    int i = blockIdx.x * blockDim.x + threadIdx.x;
    int stride = gridDim.x * blockDim.x;
    for (; i < n; i += stride) ws[i] = 0.0f;
}

// ---- per-channel counts -> reciprocal -------------------------------------
__global__ void __launch_bounds__(256) k_count(const int* __restrict__ ch,
                                               float* __restrict__ inv) {
    __shared__ int cnt[C_CH];
    if (threadIdx.x < C_CH) cnt[threadIdx.x] = 0;
    __syncthreads();
    for (int i = threadIdx.x; i < B_ROWS; i += blockDim.x)
        atomicAdd(&cnt[ch[i]], 1);
    __syncthreads();
    if (threadIdx.x < C_CH) {
        int c = cnt[threadIdx.x];
        inv[threadIdx.x] = 1.0f / (float)(c > 0 ? c : 1);
    }
}

// ---- segment-sum as one-hot GEMM on V_WMMA_F32_16X16X4_F32 ----------------
// sums(32 x 8192) = OneHot(ch)^T (32 x B) @ z (B x 8192)
// Wave owns 16 output columns; two M-tiles (ch 0-15, 16-31) share one B-tile.
// ISA 7.12.2 layouts (wave32):
//   A 16x4 : lanes 0-15 M=0-15 {K0,K1}; lanes 16-31 M=0-15 {K2,K3}
//   B 4x16 : lanes 0-15 N=0-15 {K0,K1}; lanes 16-31 N=0-15 {K2,K3}
//   C/D    : lane half selects M+8; N = lane&15; vgpr r -> M=r
// While z is in registers, also write it to the copy-half of out (each
// (b, col) is visited exactly once across the grid), so z is read only once.
__global__ void __launch_bounds__(256) k_segsum_wmma(const float* __restrict__ z,
                                                     const int*   __restrict__ ch,
                                                     float*       __restrict__ sums,
                                                     float*       __restrict__ out) {
    __shared__ int chs[BCHUNK];
    const int bbase = blockIdx.y * BCHUNK;

#if HAVE_TDM
    // Stage this K-split's 512 channel ids (2 KB) into LDS with the Tensor
    // Data Mover: 1-D tile, data_size=4B, issued by wave 0 only (TDM ignores
    // EXEC; TENSORcnt is per-wave).
    if ((threadIdx.x >> 5) == 0) {
        const unsigned long long ga =
            (unsigned long long)(const void*)(ch + bbase);
        const unsigned lds_base =
            (unsigned)(unsigned long long)(__attribute__((address_space(3))) int*)chs;
        // D# group 0: count=1 | lds_addr | global_addr[56:0] | type=2
        u32x4 g0 = { 1u,
                     lds_base,
                     (unsigned)(ga & 0xffffffffu),
                     (unsigned)((ga >> 32) & 0x01ffffffu) | (2u << 30) };
        // D# group 1: data_size=2 (4B); tensor_dim0=tile_dim0=512; stride=512
        i32x8 g1 = { (int)(2u << 16),
                     (int)((BCHUNK & 0xffff) << 16),  // tensor_dim0 @ bit 48
                     0,                               // dim0 hi / tensor_dim1 lo
                     (int)((BCHUNK & 0xffff) << 16),  // tile_dim0 @ bit 112
                     0,                               // tile_dim1/2 unused (1-D)
                     BCHUNK,                          // tensor_dim0_stride lo
                     0, 0 };
        i32x4 gz = { 0, 0, 0, 0 };
#if __has_include(<hip/amd_detail/amd_gfx1250_TDM.h>)
        i32x8 gz8 = { 0, 0, 0, 0, 0, 0, 0, 0 };
        __builtin_amdgcn_tensor_load_to_lds(g0, g1, gz, gz, gz8, 0);
#else
        __builtin_amdgcn_tensor_load_to_lds(g0, g1, gz, gz, 0);
#endif
        __builtin_amdgcn_s_wait_tensorcnt(0);
    }
#else
    for (int i = threadIdx.x; i < BCHUNK; i += blockDim.x)
        chs[i] = ch[bbase + i];
#endif
    __syncthreads();

    const int wave = threadIdx.x >> 5;
    const int lane = threadIdx.x & 31;
    const int n    = lane & 15;          // column within tile == A's M index
    const int half = lane >> 4;          // selects K pair {0,1} vs {2,3}
    const int col  = (blockIdx.x * 8 + wave) * 16 + n;

    v8f acc0 = {};   // channels 0..15
    v8f acc1 = {};   // channels 16..31

    for (int kb = 0; kb < BCHUNK; kb += 4) {
        const int r0 = kb + 2 * half;
        const int r1 = r0 + 1;
        const float z0 = z[(size_t)(bbase + r0) * ND + col];
        const float z1 = z[(size_t)(bbase + r1) * ND + col];

        // fused first-half output copy
        out[(size_t)(bbase + r0) * OUT_ROW + col] = z0;
        out[(size_t)(bbase + r1) * OUT_ROW + col] = z1;

        v2f bv; bv.x = z0; bv.y = z1;

        const int c0 = chs[r0];
        const int c1 = chs[r1];
        v2f a0, a1;
        a0.x = (c0 == n)      ? 1.0f : 0.0f;
        a0.y = (c1 == n)      ? 1.0f : 0.0f;
        a1.x = (c0 == n + 16) ? 1.0f : 0.0f;
        a1.y = (c1 == n + 16) ? 1.0f : 0.0f;

        acc0 = __builtin_amdgcn_wmma_f32_16x16x4_f32(
                   false, a0, false, bv, (short)0, acc0, false, false);
        acc1 = __builtin_amdgcn_wmma_f32_16x16x4_f32(
                   false, a1, false, bv, (short)0, acc1, false, false);
    }

    // accumulate partial tiles into the global sums table
#pragma unroll
    for (int r = 0; r < 8; ++r) {
        const int m = r + 8 * half;
        atomicAdd(&sums[(size_t)m        * ND + col], acc0[r]);
        atomicAdd(&sums[(size_t)(m + 16) * ND + col], acc1[r]);
    }
}

// ---- broadcast mean into second half of each out row (float4 stores) ------
__global__ void __launch_bounds__(256) k_broadcast(const float* __restrict__ sums,
                                                   const float* __restrict__ inv,
                                                   const int*   __restrict__ ch,
                                                   float*       __restrict__ out) {
    const int   b = blockIdx.x;
    const int   c = ch[b];
    const float s = inv[c];
    const float4* __restrict__ srow = (const float4*)(sums + (size_t)c * ND);
    float4*       __restrict__ orow = (float4*)(out + (size_t)b * OUT_ROW + ND);
    for (int i = threadIdx.x; i < ND / 4; i += blockDim.x) {
        float4 v = srow[i];
        v.x *= s; v.y *= s; v.z *= s; v.w *= s;
        orow[i] = v;
    }
}

extern "C" void kernel_launch(void* const* d_in, const int* in_sizes, int n_in,
                              void* d_out, int out_size, void* d_ws, size_t ws_size,
                              hipStream_t stream) {
    const float* z  = (const float*)d_in[0];
    const int*   ch = (const int*)d_in[1];   // harness: integer -> const int*
    float*       out = (float*)d_out;

    float* inv  = (float*)d_ws;              // 32 floats
    float* sums = inv + C_CH;                // 32 * 8192 floats (~1 MB)

    k_zero<<<256, 256, 0, stream>>>(inv, C_CH + C_CH * ND);
    k_count<<<1, 256, 0, stream>>>(ch, inv);

    dim3 grid(ND / 128, KSPLIT);             // 64 column-tile groups x 8 K-splits
    k_segsum_wmma<<<grid, 256, 0, stream>>>(z, ch, sums, out);

    k_broadcast<<<B_ROWS, 256, 0, stream>>>(sums, inv, ch, out);
}